// EnhCombHiddenLayerNN_73065983640322
// MI455X (gfx1250) — compile-verified
//
#include <hip/hip_runtime.h>
#include <hip/hip_bf16.h>

// ---------------------------------------------------------------------------
// Fused EnhCombHiddenLayerNN for MI455X (gfx1250, wave32).
// Layer1 (3->64) and Layer2 (64->3) run on V_WMMA_F32_16X16X4_F32; all
// colorimetry transcendentals run on VALU with native v_exp/v_log/v_tanh/v_rcp.
// ---------------------------------------------------------------------------

typedef float v2f __attribute__((ext_vector_type(2)));
typedef float v8f __attribute__((ext_vector_type(8)));

#define HIDDEN      64
#define HSTRIDE     66      // padded LDS stride (floats) -> conflict-free rows
#define WAVES_PER_BLOCK 8
#define BLOCK       (WAVES_PER_BLOCK * 32)

// ---- fast math helpers ----
__device__ __forceinline__ float fpow_pos(float x, float p) {   // x > 0
    return __expf(p * __logf(x));
}
__device__ __forceinline__ float ftanh_fast(float x) {
#if __has_builtin(__builtin_amdgcn_tanhf)
    return __builtin_amdgcn_tanhf(x);        // native v_tanh_f32 on CDNA5
#else
    // tanh(x) = sign(x) * (1 - e^{-2|x|}) / (1 + e^{-2|x|}); never overflows.
    float ax = fabsf(x);
    float t  = __expf(-2.0f * ax);
    float r  = (1.0f - t) * __builtin_amdgcn_rcpf(1.0f + t);  // v_rcp_f32
    return copysignf(r, x);
#endif
}
__device__ __forceinline__ float srgb_to_linear(float c) {
    float s  = fmaxf(c, 0.04045f);
    float hi = fpow_pos((s + 0.055f) * (1.0f / 1.055f), 2.4f);
    return (c <= 0.04045f) ? c * (1.0f / 12.92f) : hi;
}
__device__ __forceinline__ float linear_to_srgb(float c) {
    float s  = fmaxf(c, 0.0031308f);
    float hi = 1.055f * fpow_pos(s, 1.0f / 2.4f) - 0.055f;
    return (c <= 0.0031308f) ? 12.92f * c : hi;
}

// sRGB<->Lab constants (module constants in the reference, not inputs)
__device__ __constant__ float RGB2XYZ[9] = {
    0.412453f, 0.357580f, 0.180423f,
    0.212671f, 0.715160f, 0.072169f,
    0.019334f, 0.119193f, 0.950227f};
__device__ __constant__ float XYZ2RGB[9] = {
     3.2404542f, -1.5371385f, -0.4985314f,
    -0.9692660f,  1.8760108f,  0.0415560f,
     0.0556434f, -0.2040259f,  1.0572252f};
__device__ __constant__ float WHITE[3]     = {0.95047f, 1.0f, 1.08883f};
__device__ __constant__ float INV_WHITE[3] = {1.0f / 0.95047f, 1.0f, 1.0f / 1.08883f};
#define LAB_EPS   0.008856f
#define LAB_KAPPA 7.787f
#define F16_116   (16.0f / 116.0f)

__device__ __forceinline__ void lab2rgb_dev(float L, float a, float b,
                                            float rgb[3]) {
    float fy = (L + 16.0f) * (1.0f / 116.0f);
    float fx = fy + a * (1.0f / 500.0f);
    float fz = fy - b * (1.0f / 200.0f);
    float f[3] = {fx, fy, fz};
    float xyz[3];
    #pragma unroll
    for (int i = 0; i < 3; ++i) {
        float f3 = f[i] * f[i] * f[i];
        float t  = (f3 > LAB_EPS) ? f3 : (f[i] - F16_116) * (1.0f / LAB_KAPPA);
        xyz[i] = t * WHITE[i];
    }
    #pragma unroll
    for (int i = 0; i < 3; ++i) {
        float lin = XYZ2RGB[i * 3 + 0] * xyz[0] +
                    XYZ2RGB[i * 3 + 1] * xyz[1] +
                    XYZ2RGB[i * 3 + 2] * xyz[2];
        rgb[i] = linear_to_srgb(lin);
    }
}

__device__ __forceinline__ void rgb2lab_dev(const float rgb[3], float lab[3]) {
    float lin[3];
    #pragma unroll
    for (int i = 0; i < 3; ++i) lin[i] = srgb_to_linear(rgb[i]);
    float f[3];
    #pragma unroll
    for (int i = 0; i < 3; ++i) {
        float xyz = (RGB2XYZ[i * 3 + 0] * lin[0] +
                     RGB2XYZ[i * 3 + 1] * lin[1] +
                     RGB2XYZ[i * 3 + 2] * lin[2]) * INV_WHITE[i];
        float cb  = fpow_pos(fmaxf(xyz, LAB_EPS), 1.0f / 3.0f);
        f[i] = (xyz > LAB_EPS) ? cb : LAB_KAPPA * xyz + F16_116;
    }
    lab[0] = 116.0f * f[1] - 16.0f;
    lab[1] = 500.0f * (f[0] - f[1]);
    lab[2] = 200.0f * (f[1] - f[2]);
}

__global__ __launch_bounds__(BLOCK)
void enh_comb_nn_kernel(const float* __restrict__ x,
                        const float* __restrict__ w1, const float* __restrict__ b1,
                        const float* __restrict__ w2, const float* __restrict__ b2,
                        const float* __restrict__ wl, const float* __restrict__ bl,
                        const float* __restrict__ wc, const float* __restrict__ bc,
                        const float* __restrict__ wd, const float* __restrict__ bd,
                        const float* __restrict__ wf, const float* __restrict__ bf,
                        float* __restrict__ out, int n) {
    // per-wave LDS scratch (wave-private -> no s_barrier needed, DS is in-order)
    __shared__ __align__(16) float lds_hidden[WAVES_PER_BLOCK * 16 * HSTRIDE];
    __shared__ __align__(16) float lds_seq[WAVES_PER_BLOCK * 32 * 3];

    const int tid     = threadIdx.x;
    const int waveBlk = tid >> 5;
    const int lane    = tid & 31;
    const int lid     = lane & 15;   // N / M index inside 16-wide half
    const int half    = lane >> 4;   // which K half this lane carries

    const long long gwave = (long long)blockIdx.x * WAVES_PER_BLOCK + waveBlk;
    const long long base  = gwave * 32;
    if (base >= n) return;           // whole-wave exit keeps EXEC all-ones

    float* hid  = &lds_hidden[waveBlk * 16 * HSTRIDE];
    float* seqm = &lds_seq[waveBlk * 32 * 3];

    // ---- Layer-1 B tiles: rows = [w1 rows 0..2 ; b1] (bias rides in K=3) ----
    // B 4x16 layout: vgpr0 = {K=0 | K=2}, vgpr1 = {K=1 | K=3} across lane halves
    v2f B1[4];
    #pragma unroll
    for (int j = 0; j < 4; ++j) {
        int c = 16 * j + lid;
        B1[j][0] = (half == 0) ? w1[0 * HIDDEN + c] : w1[2 * HIDDEN + c];
        B1[j][1] = (half == 0) ? w1[1 * HIDDEN + c] : b1[c];
    }
    // ---- Layer-2 B tiles: W2[64x3] split into 16 K-slices of 4 (N>=3 -> 0) ----
    v2f B2[16];
    #pragma unroll
    for (int k = 0; k < 16; ++k) {
        int r0 = 4 * k + 2 * half;
        B2[k][0] = (lid < 3) ? w2[(r0 + 0) * 3 + lid] : 0.0f;
        B2[k][1] = (lid < 3) ? w2[(r0 + 1) * 3 + lid] : 0.0f;
    }

    // ================= two 16-row WMMA tiles per wave =================
    #pragma unroll
    for (int t = 0; t < 2; ++t) {
        long long row = base + t * 16 + lid;
        if (row >= n) row = n - 1;               // clamp; keep EXEC full

        // A 16x4 layout: vgpr0 = {K=0 | K=2}, vgpr1 = {K=1 | K=3}; K=3 is 1.0.
        // Branchless, coalesced, always in-bounds:
        //   half0 -> {x0, x1};  half1 -> {x2, 1.0}
        float va = x[row * 3 + 2 * half];
        float vb = x[row * 3 + 1];
        v2f A;
        A[0] = va;
        A[1] = half ? 1.0f : vb;

        // ---- Layer 1: hidden[16x64] = A x B1 (bias fused), 4 WMMAs ----
        v8f H[4];
        #pragma unroll
        for (int j = 0; j < 4; ++j) {
            v8f z;
            #pragma unroll
            for (int q = 0; q < 8; ++q) z[q] = 0.0f;
            H[j] = __builtin_amdgcn_wmma_f32_16x16x4_f32(
                       false, A, false, B1[j], (short)0, z, false, false);
        }

        // tanh in C/D layout, stage hidden to LDS (row-major, padded stride)
        #pragma unroll
        for (int j = 0; j < 4; ++j) {
            #pragma unroll
            for (int v = 0; v < 8; ++v) {
                int m = 8 * half + v;            // C/D: vgpr v -> M = v + 8*half
                hid[m * HSTRIDE + 16 * j + lid] = ftanh_fast(H[j][v]);
            }
        }
        __builtin_amdgcn_wave_barrier();
        asm volatile("" ::: "memory");           // DS store -> DS load ordering

        // ---- Layer 2: seq[16x3] = hidden @ W2, 16 chained WMMAs over K=64 ----
        v8f D;
        #pragma unroll
        for (int q = 0; q < 8; ++q) D[q] = 0.0f;
        #pragma unroll
        for (int k = 0; k < 16; ++k) {
            // A-layout reload: lane needs hidden[lid][4k+2h], [4k+2h+1] -> b64
            v2f A2 = *(const v2f*)&hid[lid * HSTRIDE + 4 * k + 2 * half];
            D = __builtin_amdgcn_wmma_f32_16x16x4_f32(
                    false, A2, false, B2[k], (short)0, D, false, false);
        }

        // scatter the 3 valid output columns to LDS, one row per entry
        if (lid < 3) {
            #pragma unroll
            for (int v = 0; v < 8; ++v) {
                int m = 8 * half + v;
                seqm[(t * 16 + m) * 3 + lid] = D[v];
            }
        }
        __builtin_amdgcn_wave_barrier();
        asm volatile("" ::: "memory");
    }

    // ================= elementwise phase: one row per lane =================
    long long r   = base + lane;
    bool valid    = (r < n);
    long long rr  = valid ? r : (long long)(n - 1);

    float X0 = x[rr * 3 + 0], X1 = x[rr * 3 + 1], X2 = x[rr * 3 + 2];

    float s0 = seqm[lane * 3 + 0] + b2[0];
    float s1 = seqm[lane * 3 + 1] + b2[1];
    float s2 = seqm[lane * 3 + 2] + b2[2];

    // lin = x @ w_lin + b_lin
    float lin[3];
    #pragma unroll
    for (int c = 0; c < 3; ++c)
        lin[c] = X0 * wl[0 * 3 + c] + X1 * wl[1 * 3 + c] + X2 * wl[2 * 3 + c] + bl[c];

    // final_lab = [lin, seq] @ w_comb + b_comb
    float fl[3];
    #pragma unroll
    for (int c = 0; c < 3; ++c)
        fl[c] = lin[0] * wc[0 * 3 + c] + lin[1] * wc[1 * 3 + c] + lin[2] * wc[2 * 3 + c]
              + s0     * wc[3 * 3 + c] + s1     * wc[4 * 3 + c] + s2     * wc[5 * 3 + c]
              + bc[c];

    // log-density branch
    float rgb[3];
    lab2rgb_dev(X0, X1, X2, rgb);
    float ld[3];
    #pragma unroll
    for (int i = 0; i < 3; ++i)
        ld[i] = -(__logf(rgb[i]) * 0.4342944819032518f);   // -log10
    float rgb2[3];
    #pragma unroll
    for (int c = 0; c < 3; ++c) {
        float mp = ld[0] * wd[0 * 3 + c] + ld[1] * wd[1 * 3 + c] + ld[2] * wd[2 * 3 + c] + bd[c];
        rgb2[c]  = __expf(mp * 2.302585092994046f);        // 10^mp
    }
    float back[3];
    rgb2lab_dev(rgb2, back);

    // out = [final_lab, back_to_lab] @ w_final + b_final
    #pragma unroll
    for (int c = 0; c < 3; ++c) {
        float o = fl[0]   * wf[0 * 3 + c] + fl[1]   * wf[1 * 3 + c] + fl[2]   * wf[2 * 3 + c]
                + back[0] * wf[3 * 3 + c] + back[1] * wf[4 * 3 + c] + back[2] * wf[5 * 3 + c]
                + bf[c];
        if (valid) out[r * 3 + c] = o;
    }
}

extern "C" void kernel_launch(void* const* d_in, const int* in_sizes, int n_in,
                              void* d_out, int out_size, void* d_ws, size_t ws_size,
                              hipStream_t stream) {
    const float* x  = (const float*)d_in[0];
    const float* w1 = (const float*)d_in[1];
    const float* b1 = (const float*)d_in[2];
    const float* w2 = (const float*)d_in[3];
    const float* b2 = (const float*)d_in[4];
    const float* wl = (const float*)d_in[5];
    const float* bl = (const float*)d_in[6];
    const float* wc = (const float*)d_in[7];
    const float* bc = (const float*)d_in[8];
    const float* wd = (const float*)d_in[9];
    const float* bd = (const float*)d_in[10];
    const float* wf = (const float*)d_in[11];
    const float* bf = (const float*)d_in[12];
    float* out = (float*)d_out;

    int n = in_sizes[0] / 3;                     // rows
    int blocks = (n + BLOCK - 1) / BLOCK;        // 32 rows per wave, 8 waves/block
    enh_comb_nn_kernel<<<blocks, BLOCK, 0, stream>>>(
        x, w1, b1, w2, b2, wl, bl, wc, bc, wd, bd, wf, bf, out, n);
}